// histogram_reg_25933012533653
// MI455X (gfx1250) — compile-verified
//
#include <hip/hip_runtime.h>
#include <hip/hip_bf16.h>

#define NODES 21
#define BINS  20
#define WPB   8      // waves per 256-thread block (wave32)
#define CHUNK 128    // elements per wave per tile (32 lanes x 4)

typedef float v2f __attribute__((ext_vector_type(2)));
typedef float v8f __attribute__((ext_vector_type(8)));
typedef int   v4i __attribute__((ext_vector_type(4)));

#if defined(__gfx1250__) && __has_builtin(__builtin_amdgcn_global_load_async_to_lds_b128)
#define HAVE_ASYNC 1
#else
#define HAVE_ASYNC 0
#endif

#if defined(__gfx1250__) && __has_builtin(__builtin_amdgcn_wmma_f32_16x16x4_f32)
#define HAVE_WMMA_F32X4 1
#else
#define HAVE_WMMA_F32X4 0
#endif

__device__ __forceinline__ void wait_async_le(int n) {
#if defined(__gfx1250__)
#if __has_builtin(__builtin_amdgcn_s_wait_asynccnt)
    if (n == 0) __builtin_amdgcn_s_wait_asynccnt(0);
    else        __builtin_amdgcn_s_wait_asynccnt(2);
#else
    if (n == 0) asm volatile("s_wait_asynccnt 0x0" ::: "memory");
    else        asm volatile("s_wait_asynccnt 0x2" ::: "memory");
#endif
#else
    (void)n;
#endif
}

// One sample -> 2 adjacent triangular-weighted nodes, scatter into this wave's
// private LDS sub-histogram (group g at slot offset g*32).
__device__ __forceinline__ void proc1(float x, int g, float* hg) {
    float t = x * (float)BINS;
    int   k = (int)t;
    if (k > BINS - 1) k = BINS - 1;
    if (k < 0)        k = 0;
    float f    = t - (float)k;          // weight for node k+1; (1-f) for node k
    int   base = (g ? 32 : 0) + k;
    atomicAdd(&hg[base],     1.0f - f); // ds_add_f32
    atomicAdd(&hg[base + 1], f);        // ds_add_f32
}

#if HAVE_ASYNC
__device__ __forceinline__ void issue_chunk(const float* y, const int* s,
                                            long long base, int lane,
                                            float* dY, int* dS) {
    // each lane copies its own 16B: wave moves 512B of y + 512B of s per tile
    __builtin_amdgcn_global_load_async_to_lds_b128(
        (v4i*)(y + base + (long long)lane * 4), (v4i*)(dY + lane * 4), 0, 0);
    __builtin_amdgcn_global_load_async_to_lds_b128(
        (v4i*)(s + base + (long long)lane * 4), (v4i*)(dS + lane * 4), 0, 0);
}
#endif

__global__ __launch_bounds__(256) void hist_kernel(const float* __restrict__ y,
                                                   const int* __restrict__ s,
                                                   int n, int chunks,
                                                   float* __restrict__ partials) {
    __shared__ float lhist[WPB][64];   // per-wave privatized: slot = g*32 + node
#if HAVE_ASYNC
    __shared__ __align__(16) float ldsY[2][WPB][CHUNK];
    __shared__ __align__(16) int   ldsS[2][WPB][CHUNK];
#endif
    const int tid  = threadIdx.x;
    const int wave = tid >> 5;
    const int lane = tid & 31;
    float* hg = lhist[wave];

    // wave-local init; LDS ops of one wave are in order -> no barrier needed
    hg[lane]      = 0.0f;
    hg[lane + 32] = 0.0f;

    const int W  = blockIdx.x * WPB + wave;  // global wave id
    const int TW = gridDim.x * WPB;          // total waves

#if HAVE_ASYNC
    int buf = 0;
    int c   = W;
    if (c < chunks) issue_chunk(y, s, (long long)c * CHUNK, lane,
                                ldsY[buf][wave], ldsS[buf][wave]);
    for (; c < chunks; c += TW) {
        const int next = c + TW;
        if (next < chunks)
            issue_chunk(y, s, (long long)next * CHUNK, lane,
                        ldsY[buf ^ 1][wave], ldsS[buf ^ 1][wave]);
        asm volatile("" ::: "memory");
        wait_async_le(next < chunks ? 2 : 0);  // oldest 2 asyncs (this tile) done
        asm volatile("" ::: "memory");
        const float4 xv = *reinterpret_cast<const float4*>(&ldsY[buf][wave][lane * 4]);
        const int4   sv = *reinterpret_cast<const int4*>(&ldsS[buf][wave][lane * 4]);
        proc1(xv.x, sv.x, hg);
        proc1(xv.y, sv.y, hg);
        proc1(xv.z, sv.z, hg);
        proc1(xv.w, sv.w, hg);
        buf ^= 1;
    }
#else
    for (int c = W; c < chunks; c += TW) {
        const long long base = (long long)c * CHUNK + (long long)lane * 4;
        const float4 xv = *reinterpret_cast<const float4*>(y + base);
        const int4   sv = *reinterpret_cast<const int4*>(s + base);
        proc1(xv.x, sv.x, hg);
        proc1(xv.y, sv.y, hg);
        proc1(xv.z, sv.z, hg);
        proc1(xv.w, sv.w, hg);
    }
#endif

    // tail (n % 128) handled by block 0, wave 0 with direct loads
    if (blockIdx.x == 0 && wave == 0) {
        for (int i = chunks * CHUNK + lane; i < n; i += 32)
            proc1(y[i], s[i], hg);
    }
    __syncthreads();

#if HAVE_WMMA_F32X4
    // Block reduction as ones-vector GEMM: D = ones(16x4) x H(4x16) (+C), twice
    // (waves 0-3, then 4-7) per 16-column block. Row 0 of D = column sums.
    if (tid < 32) {                      // wave 0, EXEC all ones for WMMA
        v2f a; a.x = 1.0f; a.y = 1.0f;   // A = all-ones 16x4 f32
        const int k0  = (lane < 16) ? 0 : 1;  // B VGPR0 row: K0 / K1 halves
        const int col = lane & 15;
        for (int nb = 0; nb < 4; ++nb) {
            const int ncol = nb * 16 + col;
            v8f c = {};
            v2f b;
            b.x = lhist[k0][ncol];      // B VGPR0: K = k0
            b.y = lhist[k0 + 2][ncol];  // B VGPR1: K = k0+2
            c = __builtin_amdgcn_wmma_f32_16x16x4_f32(false, a, false, b,
                                                      (short)0, c, false, false);
            b.x = lhist[k0 + 4][ncol];
            b.y = lhist[k0 + 6][ncol];
            c = __builtin_amdgcn_wmma_f32_16x16x4_f32(false, a, false, b,
                                                      (short)0, c, false, false);
            if (lane < 16) {             // D row M=0: VGPR0, lanes 0-15, N=lane
                const int slot = nb * 16 + lane;
                if ((slot & 31) < NODES) atomicAdd(&partials[slot], c[0]);
            }
        }
    }
#else
    if (tid < 64) {
        float sum = 0.0f;
        for (int w2 = 0; w2 < WPB; ++w2) sum += lhist[w2][tid];
        if ((tid & 31) < NODES) atomicAdd(&partials[tid], sum);
    }
#endif
}

__global__ void zero_ws_kernel(float* __restrict__ p) {
    p[threadIdx.x] = 0.0f;  // 64 slots
}

__global__ void finalize_kernel(const float* __restrict__ partials,
                                const float* __restrict__ pa,
                                const float* __restrict__ pb,
                                float* __restrict__ out) {
    if (threadIdx.x == 0 && blockIdx.x == 0) {
        const float fa = pa[0];
        const float fb = pb[0];
        // +1e-3 reproduces Python's int(20*0.7)==14 despite f32 20*0.7f=13.9999998
        int ab = (int)(fa * (float)BINS + 1e-3f);
        int bb = (int)(fb * (float)BINS + 1e-3f);
        if (ab < 0) ab = 0;
        if (bb > NODES) bb = NODES;
        float s0 = 0.0f, s1 = 0.0f;
        for (int i = 0; i < NODES; ++i) { s0 += partials[i]; s1 += partials[32 + i]; }
        const float inv0 = (s0 != 0.0f) ? 1.0f / s0 : 0.0f;
        const float inv1 = (s1 != 0.0f) ? 1.0f / s1 : 0.0f;
        float loss = 0.0f;
        for (int i = 0; i < NODES; ++i)
            if (i >= ab && i < bb)
                loss += fabsf(partials[i] * inv0 - partials[32 + i] * inv1);
        out[0] = loss;
    }
}

extern "C" void kernel_launch(void* const* d_in, const int* in_sizes, int n_in,
                              void* d_out, int out_size, void* d_ws, size_t ws_size,
                              hipStream_t stream) {
    const float* y  = (const float*)d_in[0];  // y_pred f32 [N]
    const int*   s  = (const int*)d_in[1];    // s       i32 [N]
    // d_in[2] = y_gt (unused by reference)
    const float* pa = (const float*)d_in[3];  // pct_a scalar
    const float* pb = (const float*)d_in[4];  // pct_b scalar
    float* out      = (float*)d_out;
    float* partials = (float*)d_ws;           // 64 f32 slots: g*32 + node

    const int n      = in_sizes[0];
    const int chunks = n / CHUNK;

    int blocks = 1024;
    int maxb = (chunks + WPB - 1) / WPB;
    if (maxb < 1) maxb = 1;
    if (blocks > maxb) blocks = maxb;

    hipLaunchKernelGGL(zero_ws_kernel, dim3(1), dim3(64), 0, stream, partials);
    hipLaunchKernelGGL(hist_kernel, dim3(blocks), dim3(256), 0, stream,
                       y, s, n, chunks, partials);
    hipLaunchKernelGGL(finalize_kernel, dim3(1), dim3(32), 0, stream,
                       partials, pa, pb, out);
}